// PyGGCN_88553635709268
// MI455X (gfx1250) — compile-verified
//
#include <hip/hip_runtime.h>
#include <hip/hip_bf16.h>

typedef __attribute__((ext_vector_type(2))) float v2f;
typedef __attribute__((ext_vector_type(8))) float v8f;
typedef __attribute__((ext_vector_type(2))) int   v2i;

#define FIN 128
#define H1  64
#define C2  16

// ---------------- utility kernels ----------------

__global__ void gcn_zero_f32(float* p, long long n) {
    long long i = (long long)blockIdx.x * blockDim.x + threadIdx.x;
    if (i < n) p[i] = 0.0f;
}

// deg[dst] += 1 for every edge (edge loaded as one b64)
__global__ void gcn_deg(const v2i* __restrict__ edges, float* __restrict__ deg, long long E) {
    long long e = (long long)blockIdx.x * blockDim.x + threadIdx.x;
    if (e >= E) return;
    v2i st = edges[e];
    __hip_atomic_fetch_add(&deg[st.y], 1.0f, __ATOMIC_RELAXED, __HIP_MEMORY_SCOPE_AGENT);
}

// d[i] = rsqrt(deg[i] + 1)   (in place)
__global__ void gcn_rsqrt(float* __restrict__ d, long long n) {
    long long i = (long long)blockIdx.x * blockDim.x + threadIdx.x;
    if (i < n) d[i] = __frsqrt_rn(d[i] + 1.0f);
}

// agg[i][f] = h[i][f]*d[i]*d[i] + bias[f]   (F compile-time -> shift/mask indexing)
template <int F>
__global__ void gcn_init_agg(const float* __restrict__ h, const float* __restrict__ d,
                             const float* __restrict__ bias, float* __restrict__ agg,
                             long long n) {
    long long idx = (long long)blockIdx.x * blockDim.x + threadIdx.x;
    if (idx >= n * F) return;
    long long i = idx / F;           // F is power of two -> shift
    int f = (int)(idx & (F - 1));
    float di = d[i];
    agg[idx] = h[idx] * di * di + bias[f];
}

// agg[dst][f] += d[src]*d[dst]*h[src][f]  over all edges, all features.
// F in {64,16}: within a 32-lane wave, idx spans 32 contiguous values, so the
// edge index e is wave-uniform (F=64) or 2 edges/wave (F=16); edge/d loads
// coalesce into single requests. Atomic lowers to global_atomic_add_f32.
template <int F>
__global__ void gcn_scatter(const v2i* __restrict__ edges, const float* __restrict__ d,
                            const float* __restrict__ h, float* __restrict__ agg,
                            long long E) {
    long long idx = (long long)blockIdx.x * blockDim.x + threadIdx.x;
    if (idx >= E * F) return;
    long long e = idx / F;           // compile-time shift
    int f = (int)(idx & (F - 1));
    v2i st = edges[e];
    float w = d[st.x] * d[st.y];
    float v = w * h[(long long)st.x * F + f];
    __hip_atomic_fetch_add(&agg[(long long)st.y * F + f], v,
                           __ATOMIC_RELAXED, __HIP_MEMORY_SCOPE_AGENT);
}

__global__ void gcn_relu(const float* __restrict__ in, float* __restrict__ out, long long n) {
    long long i = (long long)blockIdx.x * blockDim.x + threadIdx.x;
    if (i < n) out[i] = fmaxf(in[i], 0.0f);
}

// ---------------- WMMA GEMM kernels ----------------
// One wave computes one 16x16 f32 output tile with V_WMMA_F32_16X16X4_F32,
// looping K in steps of 4. A: lane m=lane&15 holds K=koff,koff+1 (koff=(lane>>4)*2)
// as a contiguous float2. B: lane n=lane&15, same K striping. D: VGPR g holds
// row g (lanes 0-15) / g+8 (lanes 16-31), col = lane&15.

// h[n x 64] = x[n x 128] @ W[128 x 64]
__global__ void __launch_bounds__(256)
gcn_gemm1_wmma(const float* __restrict__ x, const float* __restrict__ W,
               float* __restrict__ h, int nRowTiles) {
    const int lane = threadIdx.x;               // 0..31
    const int tile = blockIdx.x * blockDim.y + threadIdx.y;
    const int nTiles = nRowTiles * (H1 / 16);   // 4 column tiles
    if (tile >= nTiles) return;
    const int rowTile = tile >> 2;
    const int col0 = (tile & 3) * 16;
    const int row0 = rowTile * 16;
    const int m = lane & 15;                    // A row / B col / D col
    const int koff = (lane >> 4) << 1;          // 0 or 2

    const float* __restrict__ xrow = x + (long long)(row0 + m) * FIN;
    v8f c = {};
    #pragma unroll 4
    for (int k = 0; k < FIN; k += 4) {
        v2f a = *reinterpret_cast<const v2f*>(xrow + k + koff);  // 8B aligned
        v2f b;
        b.x = W[(long long)(k + koff) * H1 + col0 + m];
        b.y = W[(long long)(k + koff + 1) * H1 + col0 + m];
        c = __builtin_amdgcn_wmma_f32_16x16x4_f32(
                false, a, false, b, (short)0, c, false, false);
    }
    const int mBase = (lane >> 4) * 8;
    #pragma unroll
    for (int g = 0; g < 8; ++g)
        h[(long long)(row0 + mBase + g) * H1 + col0 + m] = c[g];
}

// h3[n x 16] = h2[n x 64] @ W2[64 x 16]
__global__ void __launch_bounds__(256)
gcn_gemm2_wmma(const float* __restrict__ h2, const float* __restrict__ W2,
               float* __restrict__ h3, int nRowTiles) {
    const int lane = threadIdx.x;
    const int tile = blockIdx.x * blockDim.y + threadIdx.y;  // one col tile only
    if (tile >= nRowTiles) return;
    const int row0 = tile * 16;
    const int m = lane & 15;
    const int koff = (lane >> 4) << 1;

    const float* __restrict__ hrow = h2 + (long long)(row0 + m) * H1;
    v8f c = {};
    #pragma unroll 4
    for (int k = 0; k < H1; k += 4) {
        v2f a = *reinterpret_cast<const v2f*>(hrow + k + koff);
        v2f b;
        b.x = W2[(long long)(k + koff) * C2 + m];
        b.y = W2[(long long)(k + koff + 1) * C2 + m];
        c = __builtin_amdgcn_wmma_f32_16x16x4_f32(
                false, a, false, b, (short)0, c, false, false);
    }
    const int mBase = (lane >> 4) * 8;
    #pragma unroll
    for (int g = 0; g < 8; ++g)
        h3[(long long)(row0 + mBase + g) * C2 + m] = c[g];
}

// ---------------- launch ----------------

static inline int blks(long long total, int tpb) {
    return (int)((total + tpb - 1) / tpb);
}

extern "C" void kernel_launch(void* const* d_in, const int* in_sizes, int n_in,
                              void* d_out, int out_size, void* d_ws, size_t ws_size,
                              hipStream_t stream) {
    const float* x     = (const float*)d_in[0];
    const v2i*   edges = (const v2i*)d_in[1];
    const float* W1    = (const float*)d_in[2];
    const float* b1    = (const float*)d_in[3];
    const float* W2    = (const float*)d_in[4];
    const float* b2    = (const float*)d_in[5];
    float* out = (float*)d_out;

    const long long n = in_sizes[0] / FIN;      // 100000
    const long long E = in_sizes[1] / 2;        // 3200000

    // workspace layout (floats): d | bufA (h1,then h2) | bufB (agg1) | bufC (h3)
    float* dvec = (float*)d_ws;
    float* bufA = dvec + n;
    float* bufB = bufA + n * H1;
    float* bufC = bufB + n * H1;

    const int TPB = 256;
    dim3 gblock(32, 8);   // 8 waves per block, one 16x16 tile per wave

    // degree -> d = rsqrt(deg+1)
    gcn_zero_f32<<<blks(n, TPB), TPB, 0, stream>>>(dvec, n);
    gcn_deg<<<blks(E, TPB), TPB, 0, stream>>>(edges, dvec, E);
    gcn_rsqrt<<<blks(n, TPB), TPB, 0, stream>>>(dvec, n);

    // layer 1
    const int rowTiles = (int)((n + 15) / 16);
    gcn_gemm1_wmma<<<blks((long long)rowTiles * 4, 8), gblock, 0, stream>>>(x, W1, bufA, rowTiles);
    gcn_init_agg<H1><<<blks(n * H1, TPB), TPB, 0, stream>>>(bufA, dvec, b1, bufB, n);
    gcn_scatter<H1><<<blks(E * H1, TPB), TPB, 0, stream>>>(edges, dvec, bufA, bufB, E);
    gcn_relu<<<blks(n * H1, TPB), TPB, 0, stream>>>(bufB, bufA, n * H1);

    // layer 2
    gcn_gemm2_wmma<<<blks(rowTiles, 8), gblock, 0, stream>>>(bufA, W2, bufC, rowTiles);
    gcn_init_agg<C2><<<blks(n * C2, TPB), TPB, 0, stream>>>(bufC, dvec, b2, out, n);
    gcn_scatter<C2><<<blks(E * C2, TPB), TPB, 0, stream>>>(edges, dvec, bufC, out, E);
}